// VectorQuantizer_28415503630717
// MI455X (gfx1250) — compile-verified
//
#include <hip/hip_runtime.h>
#include <hip/hip_bf16.h>

typedef float v2f __attribute__((ext_vector_type(2)));
typedef float v8f __attribute__((ext_vector_type(8)));

#define NB      32          // batch
#define NC      256         // embedding dim
#define NHW     1024        // H*W
#define NN      32768       // B*H*W rows
#define NK      1024        // codes
#define NELEM   8388608     // B*C*H*W
#define CHUNK   64          // codes per LDS chunk (64 KB)
#define NCHUNK  (NK / CHUNK)

// ---------------------------------------------------------------------------
// Kernel 1: ||e_k||^2 for each code. One wave per code.
// ---------------------------------------------------------------------------
__global__ __launch_bounds__(256) void vq_norms_kernel(const float* __restrict__ cb,
                                                       float* __restrict__ norms) {
    int code = blockIdx.x * 8 + (threadIdx.x >> 5);
    int lane = threadIdx.x & 31;
    const float* row = cb + code * NC;
    float s = 0.0f;
    #pragma unroll
    for (int i = lane; i < NC; i += 32) { float v = row[i]; s += v * v; }
    #pragma unroll
    for (int off = 16; off > 0; off >>= 1) s += __shfl_xor(s, off, 32);
    if (lane == 0) norms[code] = s;
}

// ---------------------------------------------------------------------------
// Kernel 2: distances via V_WMMA_F32_16X16X4_F32 + running argmin.
// Each wave: 16-row M-tile x 64-code N-panel (4 accumulator tiles).
// Block = 8 waves (128 rows). Grid = 256 blocks. 64 KB LDS codebook chunk.
// ---------------------------------------------------------------------------
__global__ __launch_bounds__(256) void vq_argmin_wmma_kernel(const float* __restrict__ x,
                                                             const float* __restrict__ cb,
                                                             const float* __restrict__ norms,
                                                             int*   __restrict__ idx_out_i,
                                                             float* __restrict__ idx_out_f) {
    __shared__ float ldsB[CHUNK * NC];          // 64 KB codebook chunk, [code][dim]

    const int lane = threadIdx.x & 31;
    const int wave = threadIdx.x >> 5;
    const int n0   = blockIdx.x * 128 + wave * 16;   // first row of M-tile
    const int b    = n0 >> 10;
    const int hw0  = n0 & 1023;
    // flat[n0+m][c] = x[b][c][hw0+m] at xb[c*NHW + m]
    const float* xb = x + (size_t)b * NC * NHW + hw0;

    const int m  = lane & 15;           // A row within tile
    const int kb = (lane >> 4) * 2;     // A/B k-pair base (lanes0-15: K0/1, lanes16-31: K2/3)
    const int nl = lane & 15;           // B column (code within 16-tile)

    float best[8];
    int   bidx[8];
    #pragma unroll
    for (int r = 0; r < 8; ++r) { best[r] = 3.4e38f; bidx[r] = 0; }

    for (int ch = 0; ch < NCHUNK; ++ch) {
        const int code0 = ch * CHUNK;

        // --- stage codebook chunk into LDS (coalesced float4) -------------
        __syncthreads();
        {
            const float4* src = (const float4*)(cb + (size_t)code0 * NC);
            float4*       dst = (float4*)ldsB;
            #pragma unroll
            for (int i = threadIdx.x; i < (CHUNK * NC) / 4; i += 256)
                dst[i] = src[i];
        }
        // prefetch next chunk while we compute (global_prefetch_b8)
        if (ch + 1 < NCHUNK)
            __builtin_prefetch(cb + (size_t)(code0 + CHUNK) * NC + threadIdx.x * 64, 0, 1);
        __syncthreads();

        v8f acc0 = {}; v8f acc1 = {}; v8f acc2 = {}; v8f acc3 = {};

        // --- K loop: c in steps of 4; 4 16x16 N-tiles share each A fetch --
        #pragma unroll 4
        for (int c0 = 0; c0 < NC; c0 += 4) {
            v2f a;
            a.x = xb[(size_t)(c0 + kb) * NHW + m];
            a.y = xb[(size_t)(c0 + kb + 1) * NHW + m];

            v2f b0 = *(const v2f*)&ldsB[(0 * 16 + nl) * NC + c0 + kb];
            v2f b1 = *(const v2f*)&ldsB[(1 * 16 + nl) * NC + c0 + kb];
            v2f b2 = *(const v2f*)&ldsB[(2 * 16 + nl) * NC + c0 + kb];
            v2f b3 = *(const v2f*)&ldsB[(3 * 16 + nl) * NC + c0 + kb];

            acc0 = __builtin_amdgcn_wmma_f32_16x16x4_f32(false, a, false, b0,
                                                         (short)0, acc0, false, false);
            acc1 = __builtin_amdgcn_wmma_f32_16x16x4_f32(false, a, false, b1,
                                                         (short)0, acc1, false, false);
            acc2 = __builtin_amdgcn_wmma_f32_16x16x4_f32(false, a, false, b2,
                                                         (short)0, acc2, false, false);
            acc3 = __builtin_amdgcn_wmma_f32_16x16x4_f32(false, a, false, b3,
                                                         (short)0, acc3, false, false);
        }

        // --- scores = ||e||^2 - 2*dot ; running first-min update ----------
        const int   cg0 = code0 + 0  + nl;
        const int   cg1 = code0 + 16 + nl;
        const int   cg2 = code0 + 32 + nl;
        const int   cg3 = code0 + 48 + nl;
        const float nm0 = norms[cg0];
        const float nm1 = norms[cg1];
        const float nm2 = norms[cg2];
        const float nm3 = norms[cg3];
        #pragma unroll
        for (int r = 0; r < 8; ++r) {
            float s0 = nm0 - 2.0f * acc0[r];
            float s1 = nm1 - 2.0f * acc1[r];
            float s2 = nm2 - 2.0f * acc2[r];
            float s3 = nm3 - 2.0f * acc3[r];
            if (s0 < best[r]) { best[r] = s0; bidx[r] = cg0; }
            if (s1 < best[r]) { best[r] = s1; bidx[r] = cg1; }
            if (s2 < best[r]) { best[r] = s2; bidx[r] = cg2; }
            if (s3 < best[r]) { best[r] = s3; bidx[r] = cg3; }
        }
    }

    // --- final argmin across the 16 N-lanes of each half-wave -------------
    #pragma unroll
    for (int r = 0; r < 8; ++r) {
        float v  = best[r];
        int   id = bidx[r];
        #pragma unroll
        for (int off = 1; off < 16; off <<= 1) {
            float ov = __shfl_xor(v, off, 32);
            int   oi = __shfl_xor(id, off, 32);
            if (ov < v || (ov == v && oi < id)) { v = ov; id = oi; }
        }
        if ((lane & 15) == 0) {
            // VGPR r holds rows r (lanes 0-15) and r+8 (lanes 16-31)
            int row = n0 + r + ((lane >> 4) << 3);
            idx_out_i[row] = id;
            idx_out_f[row] = (float)id;
        }
    }
}

// ---------------------------------------------------------------------------
// Kernel 3: gather codebook, straight-through output, per-block SSE partials.
// ---------------------------------------------------------------------------
__global__ __launch_bounds__(256) void vq_gather_kernel(const float* __restrict__ x,
                                                        const float* __restrict__ cb,
                                                        const int* __restrict__ idx,
                                                        float* __restrict__ q_out,
                                                        float* __restrict__ partial) {
    __shared__ float red[256];
    int gid = blockIdx.x * 256 + threadIdx.x;     // element in [B,C,H,W] layout
    int hw  = gid & 1023;
    int bc  = gid >> 10;
    int c   = bc & 255;
    int bb  = bc >> 8;
    int n   = (bb << 10) | hw;

    int   k  = idx[n];
    float q  = cb[(size_t)k * NC + c];
    float xv = x[gid];
    q_out[gid] = xv + (q - xv);                   // straight-through, IEEE-exact vs ref
    float d = q - xv;
    red[threadIdx.x] = d * d;
    __syncthreads();
    #pragma unroll
    for (int s = 128; s > 0; s >>= 1) {
        if (threadIdx.x < s) red[threadIdx.x] += red[threadIdx.x + s];
        __syncthreads();
    }
    if (threadIdx.x == 0) partial[blockIdx.x] = red[0];
}

// ---------------------------------------------------------------------------
// Kernel 4: deterministic fixed-order final loss reduction.
// ---------------------------------------------------------------------------
__global__ __launch_bounds__(256) void vq_loss_kernel(const float* __restrict__ partial,
                                                      float* __restrict__ loss_out) {
    __shared__ float red[256];
    int t = threadIdx.x;
    float s = 0.0f;
    for (int i = 0; i < 128; ++i) s += partial[t + i * 256];   // fixed order
    red[t] = s;
    __syncthreads();
    #pragma unroll
    for (int k = 128; k > 0; k >>= 1) {
        if (t < k) red[t] += red[t + k];
        __syncthreads();
    }
    if (t == 0) loss_out[0] = 1.25f * red[0] / (float)NELEM;   // q_loss + 0.25*e_loss
}

// ---------------------------------------------------------------------------
extern "C" void kernel_launch(void* const* d_in, const int* in_sizes, int n_in,
                              void* d_out, int out_size, void* d_ws, size_t ws_size,
                              hipStream_t stream) {
    const float* x  = (const float*)d_in[0];   // [32,256,32,32]
    const float* cb = (const float*)d_in[1];   // [1024,256]

    float* out      = (float*)d_out;
    float* q_out    = out;                     // 8,388,608 floats
    float* loss_out = out + NELEM;             // 1 float
    float* idx_outf = out + NELEM + 1;         // 32,768 floats (index values)

    float* norms   = (float*)d_ws;             // 1024 f32
    int*   idx_buf = (int*)(norms + NK);       // 32768 i32
    float* partial = (float*)(idx_buf + NN);   // 32768 f32

    vq_norms_kernel<<<NK / 8, 256, 0, stream>>>(cb, norms);
    vq_argmin_wmma_kernel<<<NN / 128, 256, 0, stream>>>(x, cb, norms, idx_buf, idx_outf);
    vq_gather_kernel<<<NELEM / 256, 256, 0, stream>>>(x, cb, idx_buf, q_out, partial);
    vq_loss_kernel<<<1, 256, 0, stream>>>(partial, loss_out);
}